// CGVR_69226282877395
// MI455X (gfx1250) — compile-verified
//
#include <hip/hip_runtime.h>
#include <math.h>

// Problem dims (match reference)
#define Bsz 256
#define Nn  30
#define Ll  49
#define Vv  2000
#define WDd 300
#define Dd  256
#define NBb 64
#define Rr  50

typedef __attribute__((ext_vector_type(2))) float v2f;
typedef __attribute__((ext_vector_type(8))) float v8f;

// d_out layout: outputs concatenated flat in return order
#define OUT_IMG_OFF 0
#define OUT_TGT_OFF (Bsz*NBb)                  // 16384
#define WVEC_OFF    (2*Bsz*NBb)                // 32768
#define SV_OFF      (WVEC_OFF + Bsz*Nn*WDd)    // 2336768
#define ST_OFF      (SV_OFF + Bsz*Ll)          // 2349312
#define SV2_OFF     (ST_OFF + Bsz*Nn)          // 2356992
#define KG_OFF      (SV2_OFF + Bsz*Ll)         // 2369536

// ---------------------------------------------------------------------------
// Kernel 1: KG gather (memory-bound; dominant cost, ~92 MB of HBM traffic)
// ---------------------------------------------------------------------------
__global__ __launch_bounds__(256) void kg_gather(
    const float* __restrict__ rel, const int* __restrict__ targets,
    const int* __restrict__ lengths, float* __restrict__ kg)
{
  long i = (long)blockIdx.x * blockDim.x + threadIdx.x;   // exact grid, no tail
  int r  = (int)(i % Rr);
  long t1 = i / Rr;
  int j  = (int)(t1 % Nn);
  long t2 = t1 / Nn;
  int n2 = (int)(t2 % Nn);
  int b  = (int)(t2 / Nn);
  int len = lengths[b];
  float v = 0.0f;
  if (n2 < len && j < len) {
    int ti = targets[b*Nn + n2] - 4;
    int tj = targets[b*Nn + j]  - 4;
    v = rel[((size_t)ti * Vv + tj) * Rr + r];
  }
  kg[i] = v;
}

// ---------------------------------------------------------------------------
// Kernel 2: word_vectors gather (also the GEMM A operand)
// ---------------------------------------------------------------------------
__global__ __launch_bounds__(256) void wv_gather(
    const float* __restrict__ wvs, const int* __restrict__ targets,
    float* __restrict__ outv)
{
  long i = (long)blockIdx.x * blockDim.x + threadIdx.x;
  int wd = (int)(i % WDd);
  int bn = (int)(i / WDd);
  int t  = targets[bn];
  outv[i] = wvs[(size_t)t * WDd + wd];
}

// ---------------------------------------------------------------------------
// Kernel 3: w = w_vector @ Wc^T + bc   via V_WMMA_F32_16X16X4_F32
//   M = B*N = 7680, K = 300, N = 256. 7680 tiles, 8 waves/block -> 960 blocks.
//   A frag (16x4 f32): lane m = lane&15, VGPR j holds K = 2*(lane>>4)+j.
//   C/D frag: col = lane&15, VGPR i holds row = i + 8*(lane>>4).
// ---------------------------------------------------------------------------
__global__ __launch_bounds__(256) void w_gemm(
    const float* __restrict__ wv,   // (7680, 300)
    const float* __restrict__ Wc,   // (256, 300)  (B[k][n] = Wc[n][k])
    const float* __restrict__ bc,   // (256)
    float* __restrict__ w_out)      // (7680, 256)
{
  const int lane = threadIdx.x & 31;
  const int wave = threadIdx.x >> 5;
  const int tile = blockIdx.x * 8 + wave;     // 0..7679, exact
  const int tm = tile >> 4;                    // 0..479  (M tiles)
  const int tn = tile & 15;                    // 0..15   (N tiles)

  const int m  = tm * 16 + (lane & 15);
  const int n  = tn * 16 + (lane & 15);
  const int kg = (lane >> 4) * 2;

  const float* arow = wv + (size_t)m * WDd;
  const float* brow = Wc + (size_t)n * WDd;

  v8f acc = {};
  for (int k = 0; k < WDd; k += 4) {           // 300 = 75*4, no tail
    v2f a, b;
    a.x = arow[k + kg];
    a.y = arow[k + kg + 1];
    b.x = brow[k + kg];
    b.y = brow[k + kg + 1];
    acc = __builtin_amdgcn_wmma_f32_16x16x4_f32(false, a, false, b,
                                                (short)0, acc, false, false);
  }

  const int col   = tn * 16 + (lane & 15);
  const int row0  = tm * 16 + (lane >> 4) * 8;
  const float bias = bc[col];
#pragma unroll
  for (int i = 0; i < 8; ++i)
    w_out[(size_t)(row0 + i) * Dd + col] = acc[i] + bias;
}

// ---------------------------------------------------------------------------
// Kernel 4: attention path, one batch per 256-thread block (8 waves).
//   S = softmax(img@hs^T / 16)  -> TI = S@hs -> s_v, s_v2 -> out_img head
//   Both GEMMs use f32 WMMA on padded 64x32 tiles.
// ---------------------------------------------------------------------------
__global__ __launch_bounds__(256) void attn_kernel(
    const float* __restrict__ img,  // (B,L,D)
    const float* __restrict__ hs,   // (B,N,D)
    const float* __restrict__ W1,   // (D)
    const float* __restrict__ Whi,  // (NB,D)
    const float* __restrict__ bhi,  // (NB)
    float* __restrict__ out)
{
  __shared__ float S[64 * 32];      // padded scores / attn  (8 KB)
  __shared__ float TI[Ll * Dd];     // temp_img              (~50 KB)
  __shared__ float svp[64];
  __shared__ float sv2p[64];
  __shared__ float vec[Dd];

  const int b    = blockIdx.x;
  const int tid  = threadIdx.x;
  const int lane = tid & 31;
  const int wave = tid >> 5;
  const float* imgb = img + (size_t)b * Ll * Dd;
  const float* hsb  = hs  + (size_t)b * Nn * Dd;

  // ---- Stage A: S(64x32, pad) = img(49x256) @ hs^T(256x30) * (1/16)
  {
    const int tmi = wave >> 1, tni = wave & 1;
    const int mrow = tmi * 16 + (lane & 15);   // l (pad to 64)
    const int ncol = tni * 16 + (lane & 15);   // n (pad to 32)
    const int kg   = (lane >> 4) * 2;
    const bool mok = (mrow < Ll);
    const bool nok = (ncol < Nn);
    v8f acc = {};
    for (int k = 0; k < Dd; k += 4) {
      v2f a, bb;
      a.x  = mok ? imgb[mrow * Dd + k + kg]     : 0.0f;
      a.y  = mok ? imgb[mrow * Dd + k + kg + 1] : 0.0f;
      bb.x = nok ? hsb[ncol * Dd + k + kg]      : 0.0f;
      bb.y = nok ? hsb[ncol * Dd + k + kg + 1]  : 0.0f;
      acc = __builtin_amdgcn_wmma_f32_16x16x4_f32(false, a, false, bb,
                                                  (short)0, acc, false, false);
    }
    const int scol = tni * 16 + (lane & 15);
    const int sr0  = tmi * 16 + (lane >> 4) * 8;
#pragma unroll
    for (int i = 0; i < 8; ++i)
      S[(sr0 + i) * 32 + scol] = acc[i] * 0.0625f;   // 1/sqrt(D)
  }
  __syncthreads();

  // ---- Stage B: row softmax over n (30 valid), zero padding
  if (tid < Ll) {
    float mx = -1e30f;
    for (int q = 0; q < Nn; ++q) mx = fmaxf(mx, S[tid * 32 + q]);
    float sum = 0.0f;
    for (int q = 0; q < Nn; ++q) {
      float e = __expf(S[tid * 32 + q] - mx);
      S[tid * 32 + q] = e; sum += e;
    }
    float inv = 1.0f / sum;
    for (int q = 0; q < Nn; ++q) S[tid * 32 + q] *= inv;
    S[tid * 32 + 30] = 0.0f; S[tid * 32 + 31] = 0.0f;
  } else if (tid < 64) {
    for (int q = 0; q < 32; ++q) S[tid * 32 + q] = 0.0f;  // pad rows
  }
  __syncthreads();

  // ---- Stage C: TI(49x256) = attn(49x30) @ hs(30x256), padded 64x32 K-tiles
  {
#pragma unroll 1
    for (int t = wave; t < 64; t += 8) {       // 4 Mtiles x 16 Ntiles
      const int tm2 = t >> 4, tn2 = t & 15;
      const int mrow = tm2 * 16 + (lane & 15);
      const int ncol = tn2 * 16 + (lane & 15);
      const int kg   = (lane >> 4) * 2;
      v8f acc = {};
#pragma unroll
      for (int k = 0; k < 32; k += 4) {
        v2f a, bb;
        const int k0 = k + kg, k1 = k + kg + 1;
        a.x  = S[mrow * 32 + k0];
        a.y  = S[mrow * 32 + k1];
        bb.x = (k0 < Nn) ? hsb[k0 * Dd + ncol] : 0.0f;
        bb.y = (k1 < Nn) ? hsb[k1 * Dd + ncol] : 0.0f;
        acc = __builtin_amdgcn_wmma_f32_16x16x4_f32(false, a, false, bb,
                                                    (short)0, acc, false, false);
      }
      const int sr0 = tm2 * 16 + (lane >> 4) * 8;
#pragma unroll
      for (int i = 0; i < 8; ++i) {
        const int r = sr0 + i;
        if (r < Ll) TI[r * Dd + ncol] = acc[i];
      }
    }
  }
  __syncthreads();

  // ---- Stage D: raw scores s_v = TI@W1, s_v2 = img@W1, then softmax over L
  if (tid < Ll) {
    float s1 = 0.0f, s2 = 0.0f;
    for (int c = 0; c < Dd; ++c) {
      const float w1c = W1[c];
      s1 += TI[tid * Dd + c] * w1c;
      s2 += imgb[tid * Dd + c] * w1c;
    }
    svp[tid] = s1; sv2p[tid] = s2;
  }
  __syncthreads();
  if (tid == 0) {
    float mx = -1e30f;
    for (int l = 0; l < Ll; ++l) mx = fmaxf(mx, svp[l]);
    float sum = 0.0f;
    for (int l = 0; l < Ll; ++l) { float e = __expf(svp[l] - mx); svp[l] = e; sum += e; }
    float inv = 1.0f / sum;
    for (int l = 0; l < Ll; ++l) svp[l] *= inv;
  } else if (tid == 32) {
    float mx = -1e30f;
    for (int l = 0; l < Ll; ++l) mx = fmaxf(mx, sv2p[l]);
    float sum = 0.0f;
    for (int l = 0; l < Ll; ++l) { float e = __expf(sv2p[l] - mx); sv2p[l] = e; sum += e; }
    float inv = 1.0f / sum;
    for (int l = 0; l < Ll; ++l) sv2p[l] *= inv;
  }
  __syncthreads();
  if (tid < Ll) {
    out[SV_OFF  + b * Ll + tid] = svp[tid];
    out[SV2_OFF + b * Ll + tid] = sv2p[tid];
  }

  // ---- Stage E: out_img_vec[c] = sum_l s_v[l] * img[b,l,c]
  {
    float acc = 0.0f;
    for (int l = 0; l < Ll; ++l) acc += svp[l] * imgb[l * Dd + tid];
    vec[tid] = acc;
  }
  __syncthreads();

  // ---- Stage F: out_img = sigmoid(vec @ Whi^T + bhi)
  if (tid < NBb) {
    float acc = bhi[tid];
    for (int c = 0; c < Dd; ++c) acc += vec[c] * Whi[tid * Dd + c];
    out[OUT_IMG_OFF + b * NBb + tid] = 1.0f / (1.0f + __expf(-acc));
  }
}

// ---------------------------------------------------------------------------
// Kernel 5: target path (30 rows -> scalar; reads w from workspace)
// ---------------------------------------------------------------------------
__global__ __launch_bounds__(256) void tgt_kernel(
    const float* __restrict__ hs, const float* __restrict__ tat,
    const float* __restrict__ W2, const float* __restrict__ w_ws,
    const float* __restrict__ Wht, const float* __restrict__ bht,
    float* __restrict__ out)
{
  __shared__ float stp[32];
  __shared__ float vec[Dd];
  const int b = blockIdx.x, tid = threadIdx.x;
  const float* hsb  = hs  + (size_t)b * Nn * Dd;
  const float* tatb = tat + (size_t)b * Nn * Dd;

  if (tid < Nn) {
    float s = 0.0f;
    for (int c = 0; c < Dd; ++c)
      s += tanhf(tatb[tid * Dd + c] + hsb[tid * Dd + c]) * W2[c];
    stp[tid] = s;
  }
  __syncthreads();
  if (tid == 0) {
    float mx = -1e30f;
    for (int n = 0; n < Nn; ++n) mx = fmaxf(mx, stp[n]);
    float sum = 0.0f;
    for (int n = 0; n < Nn; ++n) { float e = __expf(stp[n] - mx); stp[n] = e; sum += e; }
    float inv = 1.0f / sum;
    for (int n = 0; n < Nn; ++n) stp[n] *= inv;
  }
  __syncthreads();
  if (tid < Nn) out[ST_OFF + b * Nn + tid] = stp[tid];

  const float* wb = w_ws + (size_t)b * Nn * Dd;
  float acc = 0.0f;
  for (int n = 0; n < Nn; ++n) acc += stp[n] * wb[n * Dd + tid];
  vec[tid] = acc;
  __syncthreads();

  if (tid < NBb) {
    float a2 = bht[tid];
    for (int c = 0; c < Dd; ++c) a2 += vec[c] * Wht[tid * Dd + c];
    out[OUT_TGT_OFF + b * NBb + tid] = 1.0f / (1.0f + __expf(-a2));
  }
}

// ---------------------------------------------------------------------------
extern "C" void kernel_launch(void* const* d_in, const int* in_sizes, int n_in,
                              void* d_out, int out_size, void* d_ws, size_t ws_size,
                              hipStream_t stream) {
  (void)in_sizes; (void)n_in; (void)out_size; (void)ws_size;
  const float* rel  = (const float*)d_in[0];
  const float* wvs  = (const float*)d_in[1];
  const float* img  = (const float*)d_in[2];
  const float* hs   = (const float*)d_in[3];
  const float* tat  = (const float*)d_in[4];
  const float* Wc   = (const float*)d_in[5];
  const float* bc   = (const float*)d_in[6];
  const float* W1   = (const float*)d_in[7];
  const float* W2   = (const float*)d_in[8];
  const float* Whi  = (const float*)d_in[9];
  const float* bhi  = (const float*)d_in[10];
  const float* Wht  = (const float*)d_in[11];
  const float* bht  = (const float*)d_in[12];
  const int* targets = (const int*)d_in[13];
  const int* lengths = (const int*)d_in[14];
  float* out  = (float*)d_out;
  float* w_ws = (float*)d_ws;   // (B*N, D) = 7.86 MB scratch for `w`

  // KG: B*N*N*R = 11,520,000 elems -> 45000 exact blocks
  kg_gather<<<45000, 256, 0, stream>>>(rel, targets, lengths, out + KG_OFF);
  // w_vector: B*N*WD = 2,304,000 elems -> 9000 exact blocks
  wv_gather<<<9000, 256, 0, stream>>>(wvs, targets, out + WVEC_OFF);
  // w GEMM: 7680 WMMA tiles / 8 waves per block
  w_gemm<<<960, 256, 0, stream>>>(out + WVEC_OFF, Wc, bc, w_ws);
  // attention + image head: one batch per block
  attn_kernel<<<Bsz, 256, 0, stream>>>(img, hs, W1, Whi, bhi, out);
  // target head: one batch per block
  tgt_kernel<<<Bsz, 256, 0, stream>>>(hs, tat, W2, w_ws, Wht, bht, out);
}